// DecoderFCWithCuboic_35682588295463
// MI455X (gfx1250) — compile-verified
//
#include <hip/hip_runtime.h>
#include <hip/hip_bf16.h>
#include <math.h>

// Problem constants
#define BZ 4096
#define DD 128
#define HH 256
#define PP 2048
#define NP3 (3 * PP)   // 6144

// Output layout (flat, return order: pts, minn, cub, R)
#define OUT_PTS  0
#define OUT_MINN (BZ * 3 * PP)            // 25165824
#define OUT_CUB  (OUT_MINN + BZ)          // +4096
#define OUT_R    (OUT_CUB + BZ * 3)       // +12288

typedef __attribute__((ext_vector_type(2))) float v2f;
typedef __attribute__((ext_vector_type(8))) float v8f;

__device__ __forceinline__ float lrelu(float v) { return v > 0.0f ? v : 0.01f * v; }
__device__ __forceinline__ float sigmoidf(float v) { return 1.0f / (1.0f + expf(-v)); }

// Full-precision fp32 WMMA: D = A(16x4) * B(4x16) + C(16x16)
__device__ __forceinline__ v8f wmma_f32(v2f a, v2f b, v8f c) {
  return __builtin_amdgcn_wmma_f32_16x16x4_f32(
      /*neg_a=*/false, a, /*neg_b=*/false, b,
      /*c_mod=*/(short)0, c, /*reuse_a=*/false, /*reuse_b=*/false);
}

// ---------------------------------------------------------------------------
// 64 x N GEMM, A (64 x K, fp32) resident in LDS with stride lda,
// B (K x N, row-major) read from global (weights are L2-resident),
// epilogue: +bias, leaky-relu, store to Out (stride ldo, LDS or global).
// 8 waves per block; each wave iterates over 16x16 output tiles.
// ---------------------------------------------------------------------------
__device__ __forceinline__ void gemm64xN_lrelu(
    const float* As, int lda, int K,
    const float* __restrict__ Bg, const float* __restrict__ bias, int N,
    float* Out, int ldo, int wave, int lane)
{
  const int NT = N >> 4;          // N tiles of 16
  const int nTiles = 4 * NT;      // 4 M-tiles (64 rows)
  const int mrow  = lane & 15;
  const int khalf = (lane >> 4) << 1;   // 0 or 2

  for (int t = wave; t < nTiles; t += 8) {
    const int mt = t / NT;
    const int nt = t % NT;
    const int m = mt * 16 + mrow;
    const int n = nt * 16 + (lane & 15);

    v8f acc = {};
    const float* arow = As + m * lda;
    for (int k = 0; k < K; k += 4) {
      v2f a; a.x = arow[k + khalf]; a.y = arow[k + khalf + 1];
      v2f b; b.x = Bg[(k + khalf) * N + n]; b.y = Bg[(k + khalf + 1) * N + n];
      acc = wmma_f32(a, b, acc);
    }
    const float bn = bias[n];
    const int mbase = mt * 16 + ((lane >> 4) << 3);
#pragma unroll
    for (int r = 0; r < 8; ++r) {
      Out[(mbase + r) * ldo + n] = lrelu(acc[r] + bn);
    }
  }
}

// ---------------------------------------------------------------------------
// Kernel 0: zero the minn accumulators; write broadcast identity cuboic_R.
// ---------------------------------------------------------------------------
__global__ void init_kernel(float* __restrict__ out) {
  const int i = blockIdx.x * blockDim.x + threadIdx.x;
  if (i < BZ) out[OUT_MINN + i] = 0.0f;
  if (i < BZ * 9) {
    out[OUT_R + i] = ((i % 9) % 4 == 0) ? 1.0f : 0.0f;   // eye(3) per batch
  }
}

// ---------------------------------------------------------------------------
// Kernel 1: fused MLP trunks (both branches) for a 64-row stripe.
// ---------------------------------------------------------------------------
__global__ __launch_bounds__(256) void trunk_kernel(
    const float* __restrict__ x,
    const float* __restrict__ W1,  const float* __restrict__ b1,
    const float* __restrict__ W2,  const float* __restrict__ b2,
    const float* __restrict__ cW1, const float* __restrict__ cb1,
    const float* __restrict__ cW2, const float* __restrict__ cb2,
    const float* __restrict__ cW3, const float* __restrict__ cb3,
    float* __restrict__ h2_ws,     // [BZ][HH]
    float* __restrict__ out)
{
  extern __shared__ float smem[];
  const int LDX = DD + 4;   // 132
  const int LDH = HH + 4;   // 260
  float* Xs  = smem;                 // 64*132
  float* H1s = Xs + 64 * LDX;        // 64*260 (reused as C2s later)
  float* C1s = H1s + 64 * LDH;       // 64*260

  const int tid  = threadIdx.x;
  const int wave = tid >> 5;
  const int lane = tid & 31;
  const int row0 = blockIdx.x * 64;

  // Stage x stripe into LDS (vectorized b128 loads)
  for (int i = tid; i < 64 * (DD / 4); i += 256) {
    const int r  = i / (DD / 4);
    const int c4 = i % (DD / 4);
    const float4 v = ((const float4*)(x + (size_t)(row0 + r) * DD))[c4];
    float* dst = Xs + r * LDX + c4 * 4;
    dst[0] = v.x; dst[1] = v.y; dst[2] = v.z; dst[3] = v.w;
  }
  __syncthreads();

  // Layer 1, both branches (K = 128)
  gemm64xN_lrelu(Xs, LDX, DD, W1,  b1,  HH, H1s, LDH, wave, lane);
  gemm64xN_lrelu(Xs, LDX, DD, cW1, cb1, HH, C1s, LDH, wave, lane);
  __syncthreads();

  // Layer 2 main branch -> global workspace (K = 256)
  gemm64xN_lrelu(H1s, LDH, HH, W2, b2, HH, h2_ws + (size_t)row0 * HH, HH, wave, lane);
  __syncthreads();   // all reads of H1s done before reuse

  // Layer 2 cuboid branch -> reuse H1s as C2s
  gemm64xN_lrelu(C1s, LDH, HH, cW2, cb2, HH, H1s, LDH, wave, lane);
  __syncthreads();

  // cub head: [64][256] @ [256][3], sigmoid
  if (tid < 192) {
    const int r = tid / 3, c = tid % 3;
    float acc = cb3[c];
    const float* hrow = H1s + r * LDH;
    for (int k = 0; k < HH; ++k) acc += hrow[k] * cW3[k * 3 + c];
    out[OUT_CUB + (row0 + r) * 3 + c] = sigmoidf(acc);
  }
}

// ---------------------------------------------------------------------------
// Kernel 2: pts GEMM + fused inlier/outlier reduction.
// Block tile: 64 rows x 32 p-values x 3 coordinate groups (cols j*P + p).
// W3 tile (256 K x 96 cols) is staged in LDS with coalesced b128 loads so the
// inner loop has ZERO global loads and no cross-wave redundant B fetches.
// Each wave holds three 16x16 f32 accumulators (j = 0,1,2), so the per-point
// reduction runs in registers; pts is written exactly once (100 MB stream).
// ---------------------------------------------------------------------------
#define LDW 100   // 96 cols + 4 pad

__global__ __launch_bounds__(256) void pts_kernel(
    const float* __restrict__ h2_ws,
    const float* __restrict__ W3, const float* __restrict__ b3,
    float* __restrict__ out)
{
  extern __shared__ float smem[];
  const int LDH = HH + 4;
  float* Hs   = smem;                  // 64*260
  float* Ws   = Hs + 64 * LDH;         // 256*100 : Ws[k][j*32 + pc]
  float* cubs = Ws + HH * LDW;         // 64*3
  float* rsum = cubs + 64 * 3;         // 64 per-row partial sums

  const int tid  = threadIdx.x;
  const int wave = tid >> 5;
  const int lane = tid & 31;
  const int row0 = blockIdx.y * 64;
  const int p0   = blockIdx.x * 32;

  // Stage h2 stripe [64][256]
  for (int i = tid; i < 64 * (HH / 4); i += 256) {
    const int r  = i / (HH / 4);
    const int c4 = i % (HH / 4);
    const float4 v = ((const float4*)(h2_ws + (size_t)(row0 + r) * HH))[c4];
    float* dst = Hs + r * LDH + c4 * 4;
    dst[0] = v.x; dst[1] = v.y; dst[2] = v.z; dst[3] = v.w;
  }
  // Stage W3 tile: rows k=0..255, cols {j*PP + p0 .. +31 | j=0,1,2} -> Ws[k][j*32+pc]
  // 256 rows * 24 float4 = 6144 b128 loads, fully coalesced.
  for (int i = tid; i < HH * 24; i += 256) {
    const int k   = i / 24;
    const int rem = i % 24;
    const int j   = rem >> 3;          // 0..2
    const int c4  = rem & 7;           // 0..7
    const float4 v = ((const float4*)(W3 + (size_t)k * NP3 + j * PP + p0))[c4];
    float* dst = Ws + k * LDW + j * 32 + c4 * 4;
    dst[0] = v.x; dst[1] = v.y; dst[2] = v.z; dst[3] = v.w;
  }
  if (tid < 192) cubs[tid] = out[OUT_CUB + row0 * 3 + tid];
  if (tid < 64)  rsum[tid] = 0.0f;
  __syncthreads();

  const int mt    = wave >> 1;                 // 0..3  (row tile)
  const int nh    = wave & 1;                  // 0..1  (p half-tile)
  const int khalf = (lane >> 4) << 1;
  const int m     = mt * 16 + (lane & 15);
  const int ncol  = nh * 16 + (lane & 15);     // 0..31 within p tile
  const int p     = p0 + ncol;

  v8f acc0 = {}, acc1 = {}, acc2 = {};
  const float* arow = Hs + m * LDH;
  const float* wcol = Ws + ncol;
  for (int k = 0; k < HH; k += 4) {
    v2f a; a.x = arow[k + khalf]; a.y = arow[k + khalf + 1];
    const float* w0 = wcol + (k + khalf) * LDW;
    const float* w1 = w0 + LDW;
    v2f b0; b0.x = w0[0];  b0.y = w1[0];
    v2f b1; b1.x = w0[32]; b1.y = w1[32];
    v2f b2; b2.x = w0[64]; b2.y = w1[64];
    acc0 = wmma_f32(a, b0, acc0);
    acc1 = wmma_f32(a, b1, acc1);
    acc2 = wmma_f32(a, b2, acc2);
  }

  const int mbase = mt * 16 + ((lane >> 4) << 3);
  const float bb0 = b3[p], bb1 = b3[PP + p], bb2 = b3[2 * PP + p];
#pragma unroll
  for (int r = 0; r < 8; ++r) {
    const int mm = mbase + r;                 // stripe-local row
    const long grow = row0 + mm;
    const float c0 = cubs[mm * 3 + 0];
    const float c1 = cubs[mm * 3 + 1];
    const float c2 = cubs[mm * 3 + 2];
    const float s0 = sigmoidf(acc0[r] + bb0);
    const float s1 = sigmoidf(acc1[r] + bb1);
    const float s2 = sigmoidf(acc2[r] + bb2);
    out[grow * NP3 + 0 * PP + p] = s0;
    out[grow * NP3 + 1 * PP + p] = s1;
    out[grow * NP3 + 2 * PP + p] = s2;
    const float d0 = fabsf(c0 - s0), d1 = fabsf(c1 - s1), d2 = fabsf(c2 - s2);
    const bool i0 = (s0 <= c0), i1 = (s1 <= c1), i2 = (s2 <= c2);
    const float od = fmaxf(fmaxf(i0 ? 0.0f : fmaxf(s0, d0),
                                 i1 ? 0.0f : fmaxf(s1, d1)),
                                 i2 ? 0.0f : fmaxf(s2, d2));
    const float id = fminf(fminf(fminf(s0, d0), fminf(s1, d1)), fminf(s2, d2));
    const float contrib = (i0 && i1 && i2) ? id : od;
    atomicAdd(&rsum[mm], contrib);            // ds_add_f32
  }
  __syncthreads();
  if (tid < 64)
    atomicAdd(&out[OUT_MINN + row0 + tid], rsum[tid] * (1.0f / (float)PP));
}

// ---------------------------------------------------------------------------
extern "C" void kernel_launch(void* const* d_in, const int* in_sizes, int n_in,
                              void* d_out, int out_size, void* d_ws, size_t ws_size,
                              hipStream_t stream) {
  (void)in_sizes; (void)n_in; (void)out_size; (void)ws_size;
  const float* x   = (const float*)d_in[0];
  const float* W1  = (const float*)d_in[1];
  const float* b1  = (const float*)d_in[2];
  const float* W2  = (const float*)d_in[3];
  const float* b2  = (const float*)d_in[4];
  const float* W3  = (const float*)d_in[5];
  const float* b3  = (const float*)d_in[6];
  const float* cW1 = (const float*)d_in[7];
  const float* cb1 = (const float*)d_in[8];
  const float* cW2 = (const float*)d_in[9];
  const float* cb2 = (const float*)d_in[10];
  const float* cW3 = (const float*)d_in[11];
  const float* cb3 = (const float*)d_in[12];
  float* out   = (float*)d_out;
  float* h2_ws = (float*)d_ws;   // BZ*HH floats = 4 MB

  // 0) zero minn accumulators + identity R
  init_kernel<<<(BZ * 9 + 255) / 256, 256, 0, stream>>>(out);

  // 1) fused MLP trunks (64 stripes of 64 rows)
  const size_t smemA = (size_t)(64 * (DD + 4) + 2 * 64 * (HH + 4)) * sizeof(float); // ~167 KB
  trunk_kernel<<<BZ / 64, 256, smemA, stream>>>(
      x, W1, b1, W2, b2, cW1, cb1, cW2, cb2, cW3, cb3, h2_ws, out);

  // 2) pts GEMM + fused distance reduction (W3 tile staged in LDS)
  const size_t smemB =
      (size_t)(64 * (HH + 4) + HH * LDW + 64 * 3 + 64) * sizeof(float);             // ~168 KB
  dim3 gridB(PP / 32, BZ / 64);
  pts_kernel<<<gridB, 256, smemB, stream>>>(h2_ws, W3, b3, out);
}